// HyLoRADAUnified_14869176779251
// MI455X (gfx1250) — compile-verified
//
#include <hip/hip_runtime.h>
#include <cstddef>

// Problem constants (from the reference): B=4, S=4096, D_IN=D_OUT=4096, RANK=8
#define D_IN   4096
#define D_OUT  4096
#define RANK   8
#define NROWS  16384          // B*S
#define ROWS_PER_BLOCK 16
#define SCALING 5.65685424949238019521f   // 16 / sqrt(8)
#define EPS_F   1e-8f

typedef __attribute__((ext_vector_type(2))) float v2f;
typedef __attribute__((ext_vector_type(4))) float v4f;
typedef __attribute__((ext_vector_type(8))) float v8f;

// ---------------------------------------------------------------------------
// Kernel 1: coef[o] = (1-beta) * magnitude[o] / (||W[o,:] + SCALING*B[o,:]A||2 + eps) + beta
// One workgroup (256 threads = 8 waves) per output row. Reads base_weight once
// from HBM (64 MB total); lora_A (128 KB) stays hot in L2.
// ---------------------------------------------------------------------------
__global__ __launch_bounds__(256) void dora_coef_kernel(
    const float* __restrict__ base_weight,
    const float* __restrict__ lora_A,
    const float* __restrict__ lora_B,
    const float* __restrict__ magnitude,
    const float* __restrict__ residual_weight,
    float* __restrict__ coef)
{
    __shared__ float sB[RANK];
    __shared__ float sPart[8];

    const int o    = blockIdx.x;
    const int tid  = threadIdx.x;
    const int lane = tid & 31;
    const int wave = tid >> 5;

    if (tid < RANK) sB[tid] = lora_B[(size_t)o * RANK + tid];
    __syncthreads();

    float bl[RANK];
#pragma unroll
    for (int p = 0; p < RANK; ++p) bl[p] = sB[p];

    const float* wrow = base_weight + (size_t)o * D_IN;
    float acc = 0.0f;
    for (int c = tid; c < D_IN; c += 256) {
        float m = 0.0f;
#pragma unroll
        for (int p = 0; p < RANK; ++p)
            m = fmaf(bl[p], lora_A[(size_t)p * D_IN + c], m);
        float v = fmaf(SCALING, m, wrow[c]);
        acc = fmaf(v, v, acc);
    }
    // wave32 butterfly reduce (fixed order -> deterministic)
#pragma unroll
    for (int m = 16; m >= 1; m >>= 1) acc += __shfl_xor(acc, m, 32);
    if (lane == 0) sPart[wave] = acc;
    __syncthreads();

    if (tid == 0) {
        float s = 0.0f;
#pragma unroll
        for (int w = 0; w < 8; ++w) s += sPart[w];
        float beta = 1.0f / (1.0f + expf(-residual_weight[0]));
        float mag_scale = magnitude[o] / (sqrtf(s) + EPS_F);
        coef[o] = (1.0f - beta) * mag_scale + beta;   // (1-b)*mag_scale + b
    }
}

// ---------------------------------------------------------------------------
// Kernel 2 (fused main): per WG of 256 threads handle 16 rows of [B*S, D].
//   Step 1: lora_x_s[16][8] = SCALING * (x_rows @ lora_A^T)   (VALU, coalesced)
//   Step 2: out tile = (base_output tile + lora_x_s @ lora_B^T) * coef[n]
//           via chained V_WMMA_F32_16X16X4_F32 (rank 8 = two K=4 halves),
//           base_output fed directly as the WMMA C operand.
// ---------------------------------------------------------------------------
__global__ __launch_bounds__(256) void dora_fused_kernel(
    const float* __restrict__ x,
    const float* __restrict__ base_output,
    const float* __restrict__ lora_A,
    const float* __restrict__ lora_B,
    const float* __restrict__ coef,
    float* __restrict__ out)
{
    extern __shared__ float smem[];
    float* sA   = smem;                                  // RANK * D_IN  (128 KB)
    float* sLX  = smem + (size_t)RANK * D_IN;            // 16 * 8 scaled lora_x
    float* sRed = sLX + ROWS_PER_BLOCK * RANK;           // 8 waves * 8 ranks

    const int tid     = threadIdx.x;
    const int lane    = tid & 31;
    const int wave    = tid >> 5;
    const int rowbase = blockIdx.x * ROWS_PER_BLOCK;

    // ---- stage lora_A into LDS (coalesced b128) ----
    {
        const v4f* src = (const v4f*)lora_A;
        v4f*       dst = (v4f*)sA;
        for (int i = tid; i < (RANK * D_IN) / 4; i += 256) dst[i] = src[i];
    }
    __syncthreads();

    // ---- step 1: skinny GEMM x_rows @ A^T, fixed-order reduction ----
    for (int r = 0; r < ROWS_PER_BLOCK; ++r) {
        const float* xrow = x + (size_t)(rowbase + r) * D_IN;
        float acc[RANK];
#pragma unroll
        for (int p = 0; p < RANK; ++p) acc[p] = 0.0f;
        for (int c = tid; c < D_IN; c += 256) {
            float xv = xrow[c];
#pragma unroll
            for (int p = 0; p < RANK; ++p)
                acc[p] = fmaf(xv, sA[p * D_IN + c], acc[p]);
        }
#pragma unroll
        for (int p = 0; p < RANK; ++p) {
            float v = acc[p];
#pragma unroll
            for (int m = 16; m >= 1; m >>= 1) v += __shfl_xor(v, m, 32);
            if (lane == 0) sRed[wave * RANK + p] = v;
        }
        __syncthreads();
        if (tid < RANK) {
            float s = 0.0f;
#pragma unroll
            for (int w = 0; w < 8; ++w) s += sRed[w * RANK + tid];
            sLX[r * RANK + tid] = s * SCALING;
        }
        __syncthreads();
    }

    // ---- step 2: WMMA. ISA A-layout 16x4 f32: lanes 0-15 hold M=0-15 with
    // {VGPR0,VGPR1} = {K0,K1}; lanes 16-31 = {K2,K3}. B 4x16 symmetric.
    // C/D: VGPR j, lane l -> M = j + 8*(l>>4), N = l&15.
    const int ncol = lane & 15;
    const int hi   = lane >> 4;

    v2f a0, a1;   // A operands for rank halves [0..3] and [4..7] (loop-invariant)
    {
        const float* lxp = sLX + ncol * RANK + (hi << 1);
        a0[0] = lxp[0]; a0[1] = lxp[1];   // ranks {0,1} / {2,3}
        a1[0] = lxp[4]; a1[1] = lxp[5];   // ranks {4,5} / {6,7}
    }

    const int NT = D_OUT / 16;            // 256 column tiles, 32 per wave
    for (int i = 0; i < NT / 8; ++i) {
        const int nt   = wave + (i << 3);
        const int col0 = nt * 16 + ncol;

        // B operand: lora_B is [D_OUT, RANK] row-major -> B[k][n] = lora_B[n*8 + k]
        const float* bptr = lora_B + (size_t)col0 * RANK + (hi << 1);
        v2f b0; b0[0] = bptr[0]; b0[1] = bptr[1];
        v2f b1; b1[0] = bptr[4]; b1[1] = bptr[5];

        const float cf = coef[col0];      // per-lane: N is fixed across all 8 VGPRs

        const float* bo = base_output + (size_t)(rowbase + (hi << 3)) * D_OUT + col0;
        float*       op = out         + (size_t)(rowbase + (hi << 3)) * D_OUT + col0;

        v8f c;
#pragma unroll
        for (int j = 0; j < 8; ++j) c[j] = bo[(size_t)j * D_OUT];

        // D = lora_x_s[16x4] x lora_B^T[4x16] + C, chained over the two rank halves
        v8f d = __builtin_amdgcn_wmma_f32_16x16x4_f32(
            false, a0, false, b0, (short)0, c, false, false);
        d = __builtin_amdgcn_wmma_f32_16x16x4_f32(
            false, a1, false, b1, (short)0, d, false, false);

#pragma unroll
        for (int j = 0; j < 8; ++j) op[(size_t)j * D_OUT] = d[j] * cf;

        if (i + 1 < NT / 8)
            __builtin_prefetch(bo + 128, 0, 1);   // next tile of base_output
    }
}

// ---------------------------------------------------------------------------
extern "C" void kernel_launch(void* const* d_in, const int* in_sizes, int n_in,
                              void* d_out, int out_size, void* d_ws, size_t ws_size,
                              hipStream_t stream)
{
    const float* x        = (const float*)d_in[0];
    const float* base_out = (const float*)d_in[1];
    const float* base_w   = (const float*)d_in[2];
    const float* lora_A   = (const float*)d_in[3];
    const float* lora_B   = (const float*)d_in[4];
    const float* mag      = (const float*)d_in[5];
    const float* rw       = (const float*)d_in[6];

    float* coef = (float*)d_ws;        // 4096 floats of scratch
    float* out  = (float*)d_out;

    dora_coef_kernel<<<D_OUT, 256, 0, stream>>>(base_w, lora_A, lora_B, mag, rw, coef);

    const size_t shbytes =
        (size_t)(RANK * D_IN + ROWS_PER_BLOCK * RANK + 8 * RANK) * sizeof(float);
    dora_fused_kernel<<<NROWS / ROWS_PER_BLOCK, 256, shbytes, stream>>>(
        x, base_out, lora_A, lora_B, coef, out);
}